// StructureEncoder_27779848471415
// MI455X (gfx1250) — compile-verified
//
#include <hip/hip_runtime.h>
#include <hip/hip_bf16.h>

#define NN 50000
#define NE 800000
#define NG 256
#define D  128

typedef float v2f __attribute__((ext_vector_type(2)));
typedef float v8f __attribute__((ext_vector_type(8)));

__device__ __forceinline__ void fatomic_add(float* p, float v) {
    unsafeAtomicAdd(p, v);   // hardware global_atomic_add_f32
}

// ---------------- degree / norm ----------------
__global__ void k_deg_init(float* deg) {
    int n = blockIdx.x * blockDim.x + threadIdx.x;
    if (n < NN) deg[n] = 1.0f;                 // self loop
}

__global__ void k_deg_accum(const int* __restrict__ dst, float* __restrict__ deg) {
    int e = blockIdx.x * blockDim.x + threadIdx.x;
    if (e < NE) fatomic_add(&deg[dst[e]], 1.0f);
}

__global__ void k_dinv(const float* __restrict__ deg, float* __restrict__ dinv) {
    int n = blockIdx.x * blockDim.x + threadIdx.x;
    if (n < NN) {
        float d = deg[n];
        dinv[n] = d > 0.0f ? rsqrtf(d) : 0.0f;
    }
}

__global__ void k_edge_norm(const int* __restrict__ src, const int* __restrict__ dst,
                            const float* __restrict__ dinv, float* __restrict__ nrm) {
    int e = blockIdx.x * blockDim.x + threadIdx.x;
    if (e < NE) nrm[e] = dinv[src[e]] * dinv[dst[e]];
}

// ---------------- dense transform: out[nrows x 128] = act(in) @ W ----------------
// One 256-thread block = 8 wave32; each wave computes a 16x128 strip with
// V_WMMA_F32_16X16X4_F32 (f32 A/B, f32 accumulate), W cached in LDS.
// RELU templated so the hot loop has no per-element selects.
template <int RELU>
__global__ __launch_bounds__(256) void k_gemm_wmma(
    const float* __restrict__ in, const float* __restrict__ W,
    float* __restrict__ out, int nrows) {
    __shared__ float ldsW[128 * 129];          // stride 129 to dodge bank conflicts
    const int tid = threadIdx.x;
    for (int i = tid; i < 128 * 128; i += 256)
        ldsW[(i >> 7) * 129 + (i & 127)] = W[i];
    __syncthreads();

    const int lane  = tid & 31;
    const int wave  = tid >> 5;
    const int strip = blockIdx.x * 128 + wave * 16;
    int rowA = strip + (lane & 15);
    if (rowA >= nrows) rowA = nrows - 1;       // clamp (result discarded at store)
    const int koff = (lane >> 4) << 1;         // 0 for lanes 0-15, 2 for 16-31
    const int col  = lane & 15;

    v8f acc[8] = {};
    const float* aBase = in + (size_t)rowA * D;

    for (int k = 0; k < D; k += 4) {
        const float* ap = aBase + k + koff;
        v2f a;
        a.x = ap[0];
        a.y = ap[1];
        if (RELU) { a.x = fmaxf(a.x, 0.0f); a.y = fmaxf(a.y, 0.0f); }
        const float* b0 = &ldsW[(k + koff)     * 129 + col];
        const float* b1 = &ldsW[(k + koff + 1) * 129 + col];
#pragma unroll
        for (int j = 0; j < 8; ++j) {
            v2f b;
            b.x = b0[j * 16];
            b.y = b1[j * 16];
            acc[j] = __builtin_amdgcn_wmma_f32_16x16x4_f32(
                false, a, false, b, (short)0, acc[j], false, false);
        }
    }

    // C/D layout: VGPR r -> row strip+r (lanes 0-15) / strip+8+r (lanes 16-31)
    const int rbase = strip + (lane >> 4) * 8;
    if (strip + 16 <= nrows) {
        // fast path: whole strip in range, no per-row predication
#pragma unroll
        for (int j = 0; j < 8; ++j) {
#pragma unroll
            for (int r = 0; r < 8; ++r) {
                out[(size_t)(rbase + r) * D + j * 16 + col] = acc[j][r];
            }
        }
    } else {
#pragma unroll
        for (int j = 0; j < 8; ++j) {
#pragma unroll
            for (int r = 0; r < 8; ++r) {
                int row = rbase + r;
                if (row < nrows) out[(size_t)row * D + j * 16 + col] = acc[j][r];
            }
        }
    }
}

// ---------------- aggregation ----------------
// agg[n] = bias + h[n] * dinv[n]^2   (self-loop term; also initializes agg)
__global__ void k_agg_init(const float* __restrict__ h, const float* __restrict__ dinv,
                           const float* __restrict__ bias, float* __restrict__ agg) {
    long long i = (long long)blockIdx.x * blockDim.x + threadIdx.x;
    long long n = i >> 5;
    int lane = (int)(i & 31);
    if (n >= NN) return;
    float di = dinv[n];
    float sn = di * di;
    const float4 v = *reinterpret_cast<const float4*>(h + (size_t)n * D + lane * 4);
    const float4 b = *reinterpret_cast<const float4*>(bias + lane * 4);
    float4 o;
    o.x = b.x + v.x * sn; o.y = b.y + v.y * sn;
    o.z = b.z + v.z * sn; o.w = b.w + v.w * sn;
    *reinterpret_cast<float4*>(agg + (size_t)n * D + lane * 4) = o;
}

// one wave32 per edge; each lane handles 4 feature dims
__global__ __launch_bounds__(256) void k_edge_scatter(
    const float* __restrict__ h, const int* __restrict__ src,
    const int* __restrict__ dst, const float* __restrict__ nrmv,
    float* __restrict__ agg) {
    long long gt = (long long)blockIdx.x * blockDim.x + threadIdx.x;
    long long e  = gt >> 5;
    int lane = (int)(gt & 31);
    if (e >= NE) return;
    int s = src[e], d = dst[e];
    float nrm = nrmv[e];
    const float4 v = *reinterpret_cast<const float4*>(h + (size_t)s * D + lane * 4);
    float* o = agg + (size_t)d * D + lane * 4;
    fatomic_add(o + 0, v.x * nrm);
    fatomic_add(o + 1, v.y * nrm);
    fatomic_add(o + 2, v.z * nrm);
    fatomic_add(o + 3, v.w * nrm);
}

// ---------------- pooling ----------------
__global__ void k_pool_zero(float* sums_cnt) {                // 256*128 sums + 256 cnt
    int i = blockIdx.x * blockDim.x + threadIdx.x;
    if (i < NG * D + NG) sums_cnt[i] = 0.0f;
}

__global__ void k_pool_cnt(const int* __restrict__ batch, float* __restrict__ cnt) {
    int n = blockIdx.x * blockDim.x + threadIdx.x;
    if (n < NN) fatomic_add(&cnt[batch[n]], 1.0f);
}

__global__ void k_pool_accum(const float* __restrict__ h, const int* __restrict__ batch,
                             float* __restrict__ sums) {
    long long i = (long long)blockIdx.x * blockDim.x + threadIdx.x;
    long long n = i >> 5;
    int lane = (int)(i & 31);
    if (n >= NN) return;
    int g = batch[n];
    const float4 v = *reinterpret_cast<const float4*>(h + (size_t)n * D + lane * 4);
    float* o = sums + (size_t)g * D + lane * 4;
    fatomic_add(o + 0, v.x);
    fatomic_add(o + 1, v.y);
    fatomic_add(o + 2, v.z);
    fatomic_add(o + 3, v.w);
}

__global__ void k_pool_final(const float* __restrict__ sums, const float* __restrict__ cnt,
                             float* __restrict__ out) {
    int i = blockIdx.x * blockDim.x + threadIdx.x;
    if (i < NG * D) out[i] = sums[i] / fmaxf(cnt[i >> 7], 1.0f);
}

// ---------------- driver ----------------
extern "C" void kernel_launch(void* const* d_in, const int* in_sizes, int n_in,
                              void* d_out, int out_size, void* d_ws, size_t ws_size,
                              hipStream_t stream) {
    const float* x     = (const float*)d_in[0];
    const int*   edge  = (const int*)d_in[1];     // [2, NE]: src then dst
    const int*   batch = (const int*)d_in[2];
    const float* W1 = (const float*)d_in[3];
    const float* b1 = (const float*)d_in[4];
    const float* W2 = (const float*)d_in[5];
    const float* b2 = (const float*)d_in[6];
    const float* W3 = (const float*)d_in[7];
    const float* b3 = (const float*)d_in[8];

    const int* esrc = edge;
    const int* edst = edge + NE;

    float* ws   = (float*)d_ws;
    float* deg  = ws;                       // 50000
    float* dinv = ws + NN;                  // 50000
    float* bufH = ws + 100000;              // 50000*128  (16B aligned)
    float* bufA = bufH + (size_t)NN * D;    // 50000*128
    float* sums = bufA + (size_t)NN * D;    // 256*128
    float* cnt  = sums + NG * D;            // 256
    float* nrm  = cnt + NG;                 // 800000

    const int TB = 256;
    const int gNodes   = (NN + TB - 1) / TB;            // 196
    const int gEdges   = (NE + TB - 1) / TB;            // 3125
    const int gNode32  = (NN * 32 + TB - 1) / TB;       // 6250
    const int gEdge32  = (int)(((long long)NE * 32 + TB - 1) / TB); // 100000
    const int gGemm    = (NN + 127) / 128;              // 391

    // normalization coefficients (reused by all 3 layers)
    k_deg_init <<<gNodes, TB, 0, stream>>>(deg);
    k_deg_accum<<<gEdges, TB, 0, stream>>>(edst, deg);
    k_dinv     <<<gNodes, TB, 0, stream>>>(deg, dinv);
    k_edge_norm<<<gEdges, TB, 0, stream>>>(esrc, edst, dinv, nrm);

    // layer 1
    k_gemm_wmma<0><<<gGemm, TB, 0, stream>>>(x, W1, bufH, NN);
    k_agg_init    <<<gNode32, TB, 0, stream>>>(bufH, dinv, b1, bufA);
    k_edge_scatter<<<gEdge32, TB, 0, stream>>>(bufH, esrc, edst, nrm, bufA);

    // layer 2 (relu of layer-1 output fused into A load)
    k_gemm_wmma<1><<<gGemm, TB, 0, stream>>>(bufA, W2, bufH, NN);
    k_agg_init    <<<gNode32, TB, 0, stream>>>(bufH, dinv, b2, bufA);
    k_edge_scatter<<<gEdge32, TB, 0, stream>>>(bufH, esrc, edst, nrm, bufA);

    // layer 3
    k_gemm_wmma<1><<<gGemm, TB, 0, stream>>>(bufA, W3, bufH, NN);
    k_agg_init    <<<gNode32, TB, 0, stream>>>(bufH, dinv, b3, bufA);
    k_edge_scatter<<<gEdge32, TB, 0, stream>>>(bufH, esrc, edst, nrm, bufA);

    // global mean pool
    k_pool_zero <<<(NG * D + NG + TB - 1) / TB, TB, 0, stream>>>(sums);
    k_pool_cnt  <<<gNodes, TB, 0, stream>>>(batch, cnt);
    k_pool_accum<<<gNode32, TB, 0, stream>>>(bufA, batch, sums);
    k_pool_final<<<(NG * D + TB - 1) / TB, TB, 0, stream>>>(sums, cnt, (float*)d_out);
}